// SequenceToPair_64673617543300
// MI455X (gfx1250) — compile-verified
//
#include <hip/hip_runtime.h>
#include <hip/hip_bf16.h>

typedef float v2f __attribute__((ext_vector_type(2)));
typedef float v8f __attribute__((ext_vector_type(8)));

#define S_DIM 1024
#define INNER 32
#define PAIR  128
#define LEN   1024

// ---------------------------------------------------------------------------
// Kernel 1: LayerNorm + projection to 64 (q|k) per row.
// One block (128 threads, 4 waves) per row.
// ---------------------------------------------------------------------------
__global__ __launch_bounds__(128) void ln_proj_kernel(
    const float* __restrict__ x, const float* __restrict__ gamma,
    const float* __restrict__ beta, const float* __restrict__ pw,
    const float* __restrict__ pb, float* __restrict__ qk)
{
  __shared__ float sx[S_DIM];
  __shared__ float rsum[4], rsq[4];
  __shared__ float ph[2][64];
  const int tid = threadIdx.x;
  const int row = blockIdx.x;
  const float* xr = x + (size_t)row * S_DIM;

  float s = 0.f, sq = 0.f;
#pragma unroll
  for (int u = 0; u < 8; ++u) {
    int d = u * 128 + tid;
    float v = xr[d];
    sx[d] = v;
    s += v;
    sq += v * v;
  }
  // wave32 butterfly reduction
#pragma unroll
  for (int off = 16; off > 0; off >>= 1) {
    s  += __shfl_xor(s,  off, 32);
    sq += __shfl_xor(sq, off, 32);
  }
  if ((tid & 31) == 0) { rsum[tid >> 5] = s; rsq[tid >> 5] = sq; }
  __syncthreads();
  float ts = rsum[0] + rsum[1] + rsum[2] + rsum[3];
  float tq = rsq[0] + rsq[1] + rsq[2] + rsq[3];
  float mu  = ts * (1.0f / S_DIM);
  float var = tq * (1.0f / S_DIM) - mu * mu;
  float rs  = rsqrtf(var + 1e-5f);

#pragma unroll
  for (int u = 0; u < 8; ++u) {
    int d = u * 128 + tid;
    sx[d] = (sx[d] - mu) * rs * gamma[d] + beta[d];
  }
  __syncthreads();

  // 64 output features; two half-sums over d
  const int h = tid & 63;
  const int half = tid >> 6;
  float acc = 0.f;
  const int dbeg = half * 512;
  for (int d = dbeg; d < dbeg + 512; ++d)
    acc = fmaf(sx[d], pw[d * 64 + h], acc);
  ph[half][h] = acc;
  __syncthreads();
  if (tid < 64)
    qk[(size_t)row * 64 + tid] = ph[0][tid] + ph[1][tid] + pb[tid];
}

// ---------------------------------------------------------------------------
// Kernel 2: Qd'[j,p] = o_b[p] + sum_c q[j,c]*W2[c,p]; Kd[i,p] = sum_c k[i,c]*W2[c,p]
// W2 = o_w rows 32..63. One block per row, one thread per p.
// ---------------------------------------------------------------------------
__global__ __launch_bounds__(128) void qdkd_kernel(
    const float* __restrict__ qk, const float* __restrict__ ow,
    const float* __restrict__ ob, float* __restrict__ qdp,
    float* __restrict__ kd)
{
  const int j = blockIdx.x;
  const int p = threadIdx.x;
  const float* qrow = qk + (size_t)j * 64;
  float a = ob[p], b = 0.f;
#pragma unroll
  for (int c = 0; c < INNER; ++c) {
    float w = ow[(INNER + c) * PAIR + p];
    a = fmaf(qrow[c],         w, a);
    b = fmaf(qrow[INNER + c], w, b);
  }
  qdp[(size_t)j * PAIR + p] = a;
  kd [(size_t)j * PAIR + p] = b;
}

// ---------------------------------------------------------------------------
// Kernel 3: out[i,j,p] = (q_tile @ diag(k_i)W1)[j,p] + Qd'[j,p] - Kd[i,p]
// Grid (64 jt, 64 it); 128 threads = 4 waves; wave w owns p in [32w, 32w+32).
// Uses V_WMMA_F32_16X16X4_F32, K=32 as 8 chained WMMAs.
//
// f32 A-fragment layout (16x4 per step): lanes 0-15 hold K={kb,kb+1}=0,1;
// lanes 16-31 hold K=2,3 within each 4-wide K chunk. B mirrors with N in
// lanes. C/D: vgpr r -> row M=r (lanes 0-15) / M=r+8 (lanes 16-31), N=lane%16.
// ---------------------------------------------------------------------------
__global__ __launch_bounds__(128) void pair_kernel(
    const float* __restrict__ qk, const float* __restrict__ ow,
    const float* __restrict__ qdp, const float* __restrict__ kd,
    float* __restrict__ out)
{
  __shared__ float qs [16][INNER];   // q rows of this j-tile
  __shared__ float ksh[16][INNER];   // k rows of this i-tile
  __shared__ float qds[16][PAIR];    // Qd' tile
  __shared__ float kds[16][PAIR];    // Kd tile

  const int tid = threadIdx.x;
  const int j0 = blockIdx.x * 16;
  const int i0 = blockIdx.y * 16;

#pragma unroll
  for (int u = 0; u < 4; ++u) {
    int idx = u * 128 + tid;
    int r = idx >> 5, c = idx & 31;
    qs [r][c] = qk[(size_t)(j0 + r) * 64 + c];
    ksh[r][c] = qk[(size_t)(i0 + r) * 64 + INNER + c];
  }
#pragma unroll
  for (int u = 0; u < 16; ++u) {
    int idx = u * 128 + tid;
    int r = idx >> 7, p = idx & 127;
    qds[r][p] = qdp[(size_t)(j0 + r) * PAIR + p];
    kds[r][p] = kd [(size_t)(i0 + r) * PAIR + p];
  }
  __syncthreads();

  const int lane = tid & 31;
  const int wave = tid >> 5;
  const int m  = lane & 15;       // row / column within fragment
  const int kh = lane >> 4;       // K-half select
  const int kb = kh * 2;

  // Per-lane A fragments of q_tile for all 8 K-steps
  float aq[16];
#pragma unroll
  for (int kk = 0; kk < 8; ++kk) {
    aq[2 * kk    ] = qs[m][4 * kk + kb];
    aq[2 * kk + 1] = qs[m][4 * kk + kb + 1];
  }
  // Per-lane W1 fragments for this wave's two 16-wide p tiles (L2-resident)
  const int p0a = wave * 32;
  const int p1a = wave * 32 + 16;
  float w0[16], w1[16];
#pragma unroll
  for (int kk = 0; kk < 8; ++kk) {
    w0[2 * kk    ] = ow[(4 * kk + kb    ) * PAIR + p0a + m];
    w0[2 * kk + 1] = ow[(4 * kk + kb + 1) * PAIR + p0a + m];
    w1[2 * kk    ] = ow[(4 * kk + kb    ) * PAIR + p1a + m];
    w1[2 * kk + 1] = ow[(4 * kk + kb + 1) * PAIR + p1a + m];
  }

  for (int i = 0; i < 16; ++i) {
    v8f acc0 = {};
    v8f acc1 = {};
#pragma unroll
    for (int kk = 0; kk < 8; ++kk) {
      float kx = ksh[i][4 * kk + kb];
      float ky = ksh[i][4 * kk + kb + 1];
      v2f a;  a.x  = aq[2 * kk];      a.y  = aq[2 * kk + 1];
      v2f b0; b0.x = kx * w0[2 * kk]; b0.y = ky * w0[2 * kk + 1];
      v2f b1; b1.x = kx * w1[2 * kk]; b1.y = ky * w1[2 * kk + 1];
      acc0 = __builtin_amdgcn_wmma_f32_16x16x4_f32(
          false, a, false, b0, (short)0, acc0, false, false);
      acc1 = __builtin_amdgcn_wmma_f32_16x16x4_f32(
          false, a, false, b1, (short)0, acc1, false, false);
    }
    const int gi = i0 + i;
#pragma unroll
    for (int r = 0; r < 8; ++r) {
      int jl = r + 8 * kh;
      int pA = p0a + m, pB = p1a + m;
      float vA = acc0[r] + qds[jl][pA] - kds[i][pA];
      float vB = acc1[r] + qds[jl][pB] - kds[i][pB];
      size_t base = ((size_t)gi * LEN + (size_t)(j0 + jl)) * PAIR;
      out[base + pA] = vA;
      out[base + pB] = vB;
    }
  }
}

extern "C" void kernel_launch(void* const* d_in, const int* in_sizes, int n_in,
                              void* d_out, int out_size, void* d_ws, size_t ws_size,
                              hipStream_t stream) {
  const float* x     = (const float*)d_in[0];
  const float* gamma = (const float*)d_in[1];
  const float* beta  = (const float*)d_in[2];
  const float* pw    = (const float*)d_in[3];
  const float* pb    = (const float*)d_in[4];
  const float* ow    = (const float*)d_in[5];
  const float* ob    = (const float*)d_in[6];
  float* out = (float*)d_out;

  // workspace layout: qk[1024*64] | QdP[1024*128] | Kd[1024*128]  (1.25 MiB)
  float* qk  = (float*)d_ws;
  float* qdp = qk  + (size_t)LEN * 64;
  float* kdw = qdp + (size_t)LEN * PAIR;

  ln_proj_kernel<<<LEN, 128, 0, stream>>>(x, gamma, beta, pw, pb, qk);
  qdkd_kernel  <<<LEN, 128, 0, stream>>>(qk, ow, ob, qdp, kdw);
  pair_kernel  <<<dim3(LEN / 16, LEN / 16), 128, 0, stream>>>(qk, ow, qdp, kdw, out);
}